// GCNConv_68771016344004
// MI455X (gfx1250) — compile-verified
//
#include <hip/hip_runtime.h>

typedef __attribute__((ext_vector_type(2))) float v2f;
typedef __attribute__((ext_vector_type(8))) float v8f;

#define F_IN  128
#define F_OUT 32

// ---------------------------------------------------------------------------
// Kernel 1: Xp = X * W + b   via V_WMMA_F32_16X16X4_F32
// One wave computes a 16x32 tile of Xp (two 16x16 WMMA accumulators).
//
// Operand layouts (CDNA5 ISA 7.12.2, 32-bit data):
//   A (16x4, MxK): lane<16 -> row=lane,    v0=K0, v1=K1
//                  lane>=16-> row=lane-16, v0=K2, v1=K3
//   B (4x16, KxN): same pattern with N = lane&15
//   C/D (16x16):   VGPR j -> row = j + 8*(lane>>4), col = lane&15
// ---------------------------------------------------------------------------
__global__ __launch_bounds__(256) void gcn_linear_wmma(
    const float* __restrict__ X, const float* __restrict__ W,
    const float* __restrict__ bias, float* __restrict__ Xp, int nrows)
{
    const int lane          = threadIdx.x & 31;
    const int waveInBlock   = threadIdx.x >> 5;
    const int wavesPerBlock = blockDim.x >> 5;
    const int tile          = blockIdx.x * wavesPerBlock + waveInBlock;
    const int row0          = tile * 16;
    if (row0 >= nrows) return;           // whole wave exits -> EXEC stays all-1s for WMMA

    const int m  = lane & 15;            // A-row / B-col / C-col within tile
    const int kh = (lane >> 4) * 2;      // this lane's K pair base: 0 or 2

    const int ar = min(row0 + m, nrows - 1);      // clamped load row (generic tail safety)
    const float* __restrict__ xrow = X + (size_t)ar * F_IN;

    v8f c0 = {};   // cols 0..15
    v8f c1 = {};   // cols 16..31

#pragma unroll 8
    for (int kk = 0; kk < F_IN; kk += 4) {
        v2f a, b0, b1;
        a.x  = xrow[kk + kh + 0];
        a.y  = xrow[kk + kh + 1];
        b0.x = W[(kk + kh + 0) * F_OUT + m];
        b0.y = W[(kk + kh + 1) * F_OUT + m];
        b1.x = W[(kk + kh + 0) * F_OUT + m + 16];
        b1.y = W[(kk + kh + 1) * F_OUT + m + 16];
        // (neg_a, A, neg_b, B, c_mod, C, reuse_a, reuse_b)
        c0 = __builtin_amdgcn_wmma_f32_16x16x4_f32(false, a, false, b0,
                                                   (short)0, c0, false, false);
        c1 = __builtin_amdgcn_wmma_f32_16x16x4_f32(false, a, false, b1,
                                                   (short)0, c1, false, false);
    }

    const float bb0 = bias[m];
    const float bb1 = bias[m + 16];
    const int rbase = row0 + (lane >> 4) * 8;

#pragma unroll
    for (int j = 0; j < 8; ++j) {
        const int r = rbase + j;
        if (r < nrows) {
            Xp[(size_t)r * F_OUT + m]      = c0[j] + bb0;
            Xp[(size_t)r * F_OUT + m + 16] = c1[j] + bb1;
        }
    }
}

// ---------------------------------------------------------------------------
// Kernel 2: out[i] = sum_{e in row i} deg[e] * Xp[col[e]]
// One wave32 per node; lane == feature (F_OUT == 32 == warpSize).
// Edge (col, deg) loaded coalesced across lanes, broadcast via shuffles;
// each Xp[col] row gather is one contiguous 128B line (L2-resident).
// ---------------------------------------------------------------------------
__global__ __launch_bounds__(256) void gcn_aggregate(
    const float* __restrict__ Xp,
    const int*   __restrict__ rowptr,
    const int*   __restrict__ colidx,
    const float* __restrict__ deg,
    float* __restrict__ out, int nrows)
{
    const int lane = threadIdx.x & 31;
    const int node = (int)((blockIdx.x * blockDim.x + threadIdx.x) >> 5);
    if (node >= nrows) return;

    const int start = rowptr[node];
    const int end   = rowptr[node + 1];

    float acc = 0.0f;
    for (int e0 = start; e0 < end; e0 += 32) {
        const int eidx = e0 + lane;
        const int   c = (eidx < end) ? colidx[eidx] : 0;
        const float d = (eidx < end) ? deg[eidx]    : 0.0f;
        const int cnt = min(32, end - e0);
        for (int j = 0; j < cnt; ++j) {
            const int   cj = __shfl(c, j, 32);
            const float dj = __shfl(d, j, 32);
            acc = fmaf(dj, Xp[(size_t)cj * F_OUT + lane], acc);
        }
    }
    out[(size_t)node * F_OUT + lane] = acc;
}

// ---------------------------------------------------------------------------
extern "C" void kernel_launch(void* const* d_in, const int* in_sizes, int n_in,
                              void* d_out, int out_size, void* d_ws, size_t ws_size,
                              hipStream_t stream)
{
    const float* X      = (const float*)d_in[0];   // [N, 128]
    const float* W      = (const float*)d_in[1];   // [128, 32]
    const float* b      = (const float*)d_in[2];   // [32]
    const int*   rowptr = (const int*)  d_in[3];   // [N+1]
    const int*   colidx = (const int*)  d_in[4];   // [E]
    const float* deg    = (const float*)d_in[5];   // [E]
    float*       out    = (float*)d_out;           // [N, 32]

    const int nrows = in_sizes[3] - 1;             // len(row_pointers) - 1
    float* Xp = (float*)d_ws;                      // [N, 32] scratch (12.8 MB)

    // Pass 1: dense linear layer via WMMA. 1 wave per 16-row tile, 8 waves/block.
    const int ntiles = (nrows + 15) / 16;
    const int blocks1 = (ntiles + 7) / 8;
    gcn_linear_wmma<<<blocks1, 256, 0, stream>>>(X, W, b, Xp, nrows);

    // Pass 2: CSR aggregation. 1 wave per node, 8 nodes per 256-thread block.
    const int blocks2 = (nrows + 7) / 8;
    gcn_aggregate<<<blocks2, 256, 0, stream>>>(Xp, rowptr, colidx, deg, out, nrows);
}